// NGCF_34608846471796
// MI455X (gfx1250) — compile-verified
//
#include <hip/hip_runtime.h>
#include <hip/hip_bf16.h>

typedef __attribute__((ext_vector_type(2))) float v2f;
typedef __attribute__((ext_vector_type(8))) float v8f;

#define DIMS 64
#define LPAD 66          // padded LDS row stride (floats) to avoid bank conflicts
#define NEG_SLOPE 0.2f
#define EPS_NORM 1e-12f

// ---------------------------------------------------------------------------
// ego = concat(user_emb, item_emb)   (N x 64)
// ---------------------------------------------------------------------------
__global__ __launch_bounds__(256) void ngcf_init(const float* __restrict__ ue,
                                                 const float* __restrict__ ie,
                                                 float* __restrict__ ego,
                                                 int n_users, long long total) {
    long long idx = (long long)blockIdx.x * blockDim.x + threadIdx.x;
    if (idx >= total) return;
    long long node = idx >> 6;
    if (node < n_users) ego[idx] = ue[idx];
    else                ego[idx] = ie[idx - (long long)n_users * DIMS];
}

// ---------------------------------------------------------------------------
// SpMM scatter: side[rows[e]] += vals[e] * ego[cols[e]]
// 16 threads per edge, float4 row reads, hardware f32 atomics (L2-resident).
// ---------------------------------------------------------------------------
__global__ __launch_bounds__(256) void ngcf_spmm(const int* __restrict__ rows,
                                                 const int* __restrict__ cols,
                                                 const float* __restrict__ vals,
                                                 const float* __restrict__ ego,
                                                 float* __restrict__ side,
                                                 int nnz) {
    int e = blockIdx.x * 16 + (threadIdx.x >> 4);
    if (e >= nnz) return;
    int q = threadIdx.x & 15;
    int r = rows[e];
    int c = cols[e];
    float v = vals[e];
    const float4* src = (const float4*)(ego + (long long)c * DIMS);
    float4 x = src[q];
    float* dst = side + (long long)r * DIMS + q * 4;
    __hip_atomic_fetch_add(dst + 0, v * x.x, __ATOMIC_RELAXED, __HIP_MEMORY_SCOPE_AGENT);
    __hip_atomic_fetch_add(dst + 1, v * x.y, __ATOMIC_RELAXED, __HIP_MEMORY_SCOPE_AGENT);
    __hip_atomic_fetch_add(dst + 2, v * x.z, __ATOMIC_RELAXED, __HIP_MEMORY_SCOPE_AGENT);
    __hip_atomic_fetch_add(dst + 3, v * x.w, __ATOMIC_RELAXED, __HIP_MEMORY_SCOPE_AGENT);
}

// ---------------------------------------------------------------------------
// Dense transform per 16-node tile:
//   new_ego = leaky_relu( side@Wgc + (ego*side)@Wbi + bgc + bbi )
// 4 waves/block; each wave owns a 16-column N-tile; two chained WMMA
// accumulators (v_wmma_f32_16x16x4_f32, 16 K-steps each).
// ---------------------------------------------------------------------------
__global__ __launch_bounds__(128) void ngcf_transform(float* __restrict__ ego,
                                                      const float* __restrict__ side,
                                                      const float* __restrict__ Wgc,
                                                      const float* __restrict__ bgc,
                                                      const float* __restrict__ Wbi,
                                                      const float* __restrict__ bbi) {
    __shared__ float lds_s[16 * LPAD];   // side rows
    __shared__ float lds_b[16 * LPAD];   // (ego * side) rows
    const int node0 = blockIdx.x * 16;
    const int t = threadIdx.x;

    // Stage the A-matrices for this 16-row tile into LDS (ego consumed here,
    // so the in-place ego writeback below is race-free after the barrier).
    for (int idx = t; idx < 16 * DIMS; idx += 128) {
        int r = idx >> 6, c = idx & 63;
        float s = side[(long long)(node0 + r) * DIMS + c];
        float e = ego [(long long)(node0 + r) * DIMS + c];
        lds_s[r * LPAD + c] = s;
        lds_b[r * LPAD + c] = e * s;
    }
    __syncthreads();

    const int lane = t & 31;
    const int ln   = lane & 15;          // row (A) / column-in-tile (B,D)
    const int hi   = lane >> 4;          // K half selector for A/B fragments
    const int j0   = (t >> 5) * 16;      // this wave's N-tile base column

    v8f acc_gc = {};
    v8f acc_bi = {};
    #pragma unroll
    for (int kk = 0; kk < 16; ++kk) {
        const int kb = 4 * kk + 2 * hi;  // base K index held by this lane
        v2f a1, a2, b1, b2;
        // A 16x4 fragment layout: lane<16 -> K={kb,kb+1}, lane>=16 -> K={kb,kb+1} w/ kb+=2
        a1.x = lds_s[ln * LPAD + kb];     a1.y = lds_s[ln * LPAD + kb + 1];
        a2.x = lds_b[ln * LPAD + kb];     a2.y = lds_b[ln * LPAD + kb + 1];
        // B 4x16 fragment (transpose of A layout): N = j0+ln, same K pair
        b1.x = Wgc[kb * DIMS + j0 + ln];  b1.y = Wgc[(kb + 1) * DIMS + j0 + ln];
        b2.x = Wbi[kb * DIMS + j0 + ln];  b2.y = Wbi[(kb + 1) * DIMS + j0 + ln];
        acc_gc = __builtin_amdgcn_wmma_f32_16x16x4_f32(false, a1, false, b1,
                                                       (short)0, acc_gc, false, false);
        acc_bi = __builtin_amdgcn_wmma_f32_16x16x4_f32(false, a2, false, b2,
                                                       (short)0, acc_bi, false, false);
    }

    const int col  = j0 + ln;
    const float bias = bgc[col] + bbi[col];
    #pragma unroll
    for (int jj = 0; jj < 8; ++jj) {
        // C/D layout: VGPR jj, lane<16 -> M=jj ; lane>=16 -> M=jj+8
        float v = acc_gc[jj] + acc_bi[jj] + bias;
        v = (v >= 0.0f) ? v : NEG_SLOPE * v;
        int m = jj + 8 * hi;
        ego[(long long)(node0 + m) * DIMS + col] = v;
    }
}

// ---------------------------------------------------------------------------
// Gather 3*B rows (u from users, i/j from items) of the current ego into
// output columns [64*layer, 64*layer+64), optionally L2-row-normalized.
// One wave per output row; shuffle reduction for the norm.
// ---------------------------------------------------------------------------
__global__ __launch_bounds__(256) void ngcf_gather(const float* __restrict__ src,
                                                   const int* __restrict__ u,
                                                   const int* __restrict__ ii,
                                                   const int* __restrict__ jj,
                                                   float* __restrict__ out,
                                                   int Bn, int n_users,
                                                   int layer, int normalize) {
    int wrow = blockIdx.x * (blockDim.x >> 5) + (threadIdx.x >> 5);
    int lane = threadIdx.x & 31;
    if (wrow >= 3 * Bn) return;
    int which = wrow / Bn;
    int b     = wrow - which * Bn;
    int node  = (which == 0) ? u[b]
              : n_users + ((which == 1) ? ii[b] : jj[b]);
    const float* p = src + (long long)node * DIMS;
    float x0 = p[lane * 2];
    float x1 = p[lane * 2 + 1];
    if (normalize) {
        float ss = x0 * x0 + x1 * x1;
        #pragma unroll
        for (int off = 16; off > 0; off >>= 1)
            ss += __shfl_xor(ss, off, 32);
        float inv = 1.0f / fmaxf(sqrtf(ss), EPS_NORM);
        x0 *= inv;
        x1 *= inv;
    }
    float* o = out + (long long)which * Bn * 256 + (long long)b * 256 + layer * DIMS;
    o[lane * 2]     = x0;
    o[lane * 2 + 1] = x1;
}

// ---------------------------------------------------------------------------
extern "C" void kernel_launch(void* const* d_in, const int* in_sizes, int n_in,
                              void* d_out, int out_size, void* d_ws, size_t ws_size,
                              hipStream_t stream) {
    const int*   rows = (const int*)  d_in[0];
    const int*   cols = (const int*)  d_in[1];
    const float* vals = (const float*)d_in[2];
    const float* ue   = (const float*)d_in[3];
    const float* ie   = (const float*)d_in[4];
    const float* Wgc  = (const float*)d_in[5];
    const float* bgc  = (const float*)d_in[6];
    const float* Wbi  = (const float*)d_in[7];
    const float* bbi  = (const float*)d_in[8];
    const int*   u    = (const int*)  d_in[9];
    const int*   iidx = (const int*)  d_in[10];
    const int*   jidx = (const int*)  d_in[11];

    const int nnz     = in_sizes[0];
    const int n_users = in_sizes[3] / DIMS;
    const int n_items = in_sizes[4] / DIMS;
    const int N       = n_users + n_items;
    const int Bn      = in_sizes[9];
    float* out = (float*)d_out;

    float* ego  = (float*)d_ws;                    // N x 64
    float* side = ego + (long long)N * DIMS;       // N x 64

    const long long tot = (long long)N * DIMS;

    ngcf_init<<<(unsigned)((tot + 255) / 256), 256, 0, stream>>>(ue, ie, ego, n_users, tot);

    const int gblocks = (3 * Bn + 7) / 8;          // 8 waves per 256-thread block
    ngcf_gather<<<gblocks, 256, 0, stream>>>(ego, u, iidx, jidx, out, Bn, n_users, 0, 0);

    for (int k = 0; k < 3; ++k) {
        hipMemsetAsync(side, 0, tot * sizeof(float), stream);
        ngcf_spmm<<<(nnz + 15) / 16, 256, 0, stream>>>(rows, cols, vals, ego, side, nnz);
        ngcf_transform<<<N / 16, 128, 0, stream>>>(ego, side,
                                                   Wgc + (long long)k * DIMS * DIMS,
                                                   bgc + (long long)k * DIMS,
                                                   Wbi + (long long)k * DIMS * DIMS,
                                                   bbi + (long long)k * DIMS);
        ngcf_gather<<<gblocks, 256, 0, stream>>>(ego, u, iidx, jidx, out, Bn, n_users, k + 1, 1);
    }
}